// CSTR_91319594647919
// MI455X (gfx1250) — compile-verified
//
#include <hip/hip_runtime.h>
#include <hip/hip_bf16.h>
#include <stdint.h>

// ---------------------------------------------------------------------------
// CSTR batched rollout: B=4096 samples, T=2047 sequential steps, 2-state ODE.
// One thread per sample. w[B][2][2047] staged HBM -> LDS via TDM
// (tensor_load_to_lds), triple-buffered per wave (prefetch distance 2),
// TDM pad inserts 1 dword per 16-dword row => bank-conflict-free LDS reads.
// tanh/sigmoid via hardware v_tanh_f32 (gfx1250 TRANS op), paired to
// amortize the TRANS-hazard v_nop.
// ---------------------------------------------------------------------------

typedef __attribute__((ext_vector_type(4))) unsigned int u32x4;
typedef __attribute__((ext_vector_type(8))) int          i32x8;
typedef __attribute__((ext_vector_type(4))) int          i32x4;

#define HSTEP   0.01f
#define TSTEPS  2047
#define TT      16                 // time steps per tile
#define NTILES  128                // ceil(2047/16)
#define WROWS   64                 // rows per wave tile: 32 samples * 2 ch
#define ROWDW   17                 // 16 data dwords + 1 pad dword (TDM pad)
#define BUFDW   (WROWS * ROWDW)    // 1088 dwords per buffer per wave
#define NBUF    3                  // triple buffer
#define WAVEDW  (NBUF * BUFDW)     // 3264 dwords per wave
#define NWAVES  2
#define BLOCK   (NWAVES * 32)

// Single hardware tanh; trailing v_nop satisfies the TRANS hazard rule.
__device__ __forceinline__ float htanh(float x) {
    float r;
    asm volatile("v_tanh_f32 %0, %1\n\tv_nop" : "=v"(r) : "v"(x));
    return r;
}

// Two independent hardware tanh: the second trans op is the "1 independent
// op" covering the first, so one v_nop covers both results.
__device__ __forceinline__ void htanh2(float& ra, float& rb, float xa, float xb) {
    asm volatile("v_tanh_f32 %0, %2\n\t"
                 "v_tanh_f32 %1, %3\n\t"
                 "v_nop"
                 : "=&v"(ra), "=&v"(rb)
                 : "v"(xa), "v"(xb));
}

// Issue one TDM 2D tile load: 64 rows x TT floats, row stride 2047 floats,
// into LDS at lds_byte with 1 pad dword per 16-dword row.
__device__ __forceinline__ void tdm_issue(const float* gbase, uint32_t lds_byte,
                                          int t0) {
    uint64_t ga = (uint64_t)(uintptr_t)gbase + (uint64_t)(uint32_t)t0 * 4u;

    u32x4 g0;
    g0[0] = 1u;                                   // count=1, user descriptor
    g0[1] = lds_byte;                             // lds_addr (bytes)
    g0[2] = (uint32_t)ga;                         // global_addr[31:0]
    g0[3] = ((uint32_t)(ga >> 32) & 0x01FFFFFFu)  // global_addr[56:32]
            | 0x80000000u;                        // type=2 ("image")

    i32x8 g1;
    g1[0] = (int)0x00D20000u;       // data_size=4B, pad_enable=1,
                                    // pad_interval=3 (16 dw), pad_amount=0 (1 dw)
    g1[1] = (int)(2047u << 16);     // tensor_dim0[15:0]=2047
    g1[2] = (int)(64u << 16);       // tensor_dim0[31:16]=0 | tensor_dim1=64
    g1[3] = (int)((unsigned)TT << 16); // tile_dim0=16
    g1[4] = WROWS;                  // tile_dim1=64, tile_dim2=0
    g1[5] = 2047;                   // tensor_dim0_stride[31:0]
    g1[6] = 0;                      // stride hi / dim1_stride lo (2D: unused)
    g1[7] = 0;

    i32x4 z4 = {0, 0, 0, 0};        // groups 2/3 unused (<=2D tensor)
    i32x8 z8 = {0, 0, 0, 0, 0, 0, 0, 0};
    __builtin_amdgcn_tensor_load_to_lds(g0, g1, z4, z4, z8, 0);
}

__global__ void __launch_bounds__(BLOCK)
cstr_rollout(const float* __restrict__ w,   // [B][2][2047]
             const float* __restrict__ K,   // [1][2]
             const float* __restrict__ L,   // [4][4]
             const float* __restrict__ M,   // [1][4]
             const float* __restrict__ Mo,  // [1][1]
             float* __restrict__ out) {     // [B]
    __shared__ float smem[NWAVES * WAVEDW];

    const int tid  = threadIdx.x;
    const int lane = tid & 31;
    const int wav  = __builtin_amdgcn_readfirstlane(tid >> 5);
    const int gs0  = blockIdx.x * BLOCK + wav * 32;  // first sample of wave
    const int s    = gs0 + lane;                     // this thread's sample

    // Wave-uniform slab base: rows of length 2047, row r = (sample-gs0)*2+ch.
    const float* gbase = w + (size_t)gs0 * 2u * 2047u;

    // LDS byte address of this wave's region (low 32 bits of generic addr ==
    // LDS byte offset per the gfx1250 aperture mapping).
    uint32_t lds0 = (uint32_t)(uintptr_t)&smem[wav * WAVEDW];
    lds0 = (uint32_t)__builtin_amdgcn_readfirstlane((int)lds0);

    // Small uniform parameter loads (scalarized by the compiler).
    const float K0 = K[0], K1 = K[1];
    const float L00=L[0], L01=L[1], L02=L[2], L03=L[3];
    const float L10=L[4], L11=L[5], L12=L[6], L13=L[7];
    const float L20=L[8], L21=L[9], L22=L[10], L23=L[11];
    const float L30=L[12], L31=L[13], L32=L[14], L33=L[15];
    const float M0v=M[0], M1v=M[1], M2v=M[2], M3v=M[3];
    const float Mo0=Mo[0];

    // Prologue: prefetch tiles 0 and 1.
    tdm_issue(gbase, lds0 + 0u * (BUFDW * 4u), 0);
    tdm_issue(gbase, lds0 + 1u * (BUFDW * 4u), TT);

    float x1 = 1.f, x2 = 0.f;       // x
    float xh1 = 1.f, xh2 = 0.f;     // xhat_prev
    float up = 0.f;                 // u_prev
    float J = 0.f;

    int bufk = 0;                    // k % NBUF
    for (int k = 0; k < NTILES; ++k) {
        if (k + 2 < NTILES) {
            int b2 = bufk + 2; if (b2 >= NBUF) b2 -= NBUF;
            tdm_issue(gbase, lds0 + (uint32_t)b2 * (BUFDW * 4u), (k + 2) * TT);
            __builtin_amdgcn_s_wait_tensorcnt(2);  // tile k landed (in-order)
        } else if (k + 1 < NTILES) {
            __builtin_amdgcn_s_wait_tensorcnt(1);
        } else {
            __builtin_amdgcn_s_wait_tensorcnt(0);
        }
        asm volatile("" ::: "memory");

        // Conflict-free LDS rows: lane l -> dwords 34l (c=0), 34l+17 (c=1).
        const float* p0 = &smem[wav * WAVEDW + bufk * BUFDW + lane * (2 * ROWDW)];
        const float* p1 = p0 + ROWDW;

        const int tbase = k * TT;
        const int xmax  = (tbase + TT <= TSTEPS) ? TT : (TSTEPS - tbase);

        for (int xi = 0; xi < xmax; ++xi) {
            const int tt = tbase + xi;
            const float w1 = p0[xi];
            const float w2 = p1[xi];

            // fprev = f(xhat_prev, u_prev)  (RK sums collapsed exactly)
            float th1, ta, tb, tc;
            htanh2(th1, ta, xh1, xh2);
            htanh2(tb, tc, xh2 + 0.005f, xh2 + 0.01f);
            const float a1  = (ta + 4.f * tb + tc) * (1.f / 6.f) - xh1 + up;
            const float a2  = -xh2 - 0.005f - th1 + 0.5f * up;
            const float f1  = fmaf(HSTEP, a1, xh1);
            const float f2  = fmaf(HSTEP, a2, xh2);

            // Independent tanh for the x-update f(x,u): only depend on x,
            // so they overlap with the phi/sigmoid chain below.
            float sx1, sa, sb, sc;
            htanh2(sx1, sa, x1, x2);
            htanh2(sb, sc, x2 + 0.005f, x2 + 0.01f);

            // phi = rx L rx + M rx + Mo,  rx = (x1, x2, f1, f2)
            const float q0 = fmaf(L00,x1, fmaf(L01,x2, fmaf(L02,f1, fmaf(L03,f2, M0v))));
            const float q1 = fmaf(L10,x1, fmaf(L11,x2, fmaf(L12,f1, fmaf(L13,f2, M1v))));
            const float q2 = fmaf(L20,x1, fmaf(L21,x2, fmaf(L22,f1, fmaf(L23,f2, M2v))));
            const float q3 = fmaf(L30,x1, fmaf(L31,x2, fmaf(L32,f1, fmaf(L33,f2, M3v))));
            const float phi = fmaf(x1,q0, fmaf(x2,q1, fmaf(f1,q2, fmaf(f2,q3, Mo0))));

            // sigmoid(phi) = 0.5 + 0.5*tanh(phi/2); forced to 1 at t==0.
            float delta = fmaf(0.5f, htanh(0.5f * phi), 0.5f);
            if (tt == 0) delta = 1.f;

            // xhat = f + delta*(x - f);  u = K @ xhat
            const float nh1 = fmaf(delta, x1 - f1, f1);
            const float nh2 = fmaf(delta, x2 - f2, f2);
            const float u   = fmaf(K0, nh1, K1 * nh2);

            // stage cost over t in [0, T-2]: |x|^2 + (K.x)^2 + delta
            if (tt < TSTEPS - 1) {
                const float us = fmaf(K0, x1, K1 * x2);
                J += fmaf(x1, x1, fmaf(x2, x2, fmaf(us, us, delta)));
            }

            // x_next = f(x, u) + w_t
            const float b1 = (sa + 4.f * sb + sc) * (1.f / 6.f) - x1 + u;
            const float b2 = -x2 - 0.005f - sx1 + 0.5f * u;
            x1 = fmaf(HSTEP, b1, x1) + w1;
            x2 = fmaf(HSTEP, b2, x2) + w2;

            xh1 = nh1; xh2 = nh2; up = u;
        }

        ++bufk; if (bufk == NBUF) bufk = 0;
    }

    // final cost: x_T @ (10 I) @ x_T
    out[s] = fmaf(10.f, fmaf(x1, x1, x2 * x2), J);
}

extern "C" void kernel_launch(void* const* d_in, const int* in_sizes, int n_in,
                              void* d_out, int out_size, void* d_ws, size_t ws_size,
                              hipStream_t stream) {
    (void)n_in; (void)d_ws; (void)ws_size; (void)in_sizes;
    const float* w  = (const float*)d_in[0];
    const float* K  = (const float*)d_in[1];
    const float* L  = (const float*)d_in[2];
    const float* M  = (const float*)d_in[3];
    const float* Mo = (const float*)d_in[4];
    float* out = (float*)d_out;

    const int B = out_size;              // 4096 samples
    const int blocks = (B + BLOCK - 1) / BLOCK;
    cstr_rollout<<<blocks, BLOCK, 0, stream>>>(w, K, L, M, Mo, out);
}